// xLSTMLayer_58067957842090
// MI455X (gfx1250) — compile-verified
//
#include <hip/hip_runtime.h>
#include <math.h>

// ---------------------------------------------------------------------------
// xLSTM layer for MI455X (gfx1250, wave32, WMMA)
//   B=16, T=512, H=128
//   Kernel 1: 6 fused projections via V_WMMA_F32_16X16X4_F32 (exact f32)
//   Kernel 2: chunkwise-parallel mLSTM scan (chunk L=16):
//               P  = C0 @ Q^T                (128x16x128  WMMA)
//               S  = K  @ Q^T                (16x16x128   WMMA)
//               C' = diag(G15) C0 + Wc @ K   (128x128x16  WMMA)
//             C resident in LDS; per-row decays/weights on VALU;
//             next chunk's gates software-prefetched into VGPRs.
//   Kernel 3: residual + LayerNorm, one wave per row
// ---------------------------------------------------------------------------

typedef float v2f __attribute__((ext_vector_type(2)));
typedef float v8f __attribute__((ext_vector_type(8)));

#define BB   16
#define TT   512
#define HH   128
#define MM   (BB * TT)          // 8192 rows
#define NEL  (BB * TT * HH)     // elements per (B,T,H) tensor
#define SCALE 0.08838834764831845f  // 1/sqrt(128)
#define LN_EPS 1e-5f
#define LCH  16                 // chunk length
#define NCH  (TT / LCH)         // 32 chunks

// ---------------------------------------------------------------------------
// Kernel 1: projections.
// ---------------------------------------------------------------------------
__global__ __launch_bounds__(128) void proj_kernel(
    const float* __restrict__ x,
    const float* __restrict__ Wq, const float* __restrict__ Wk,
    const float* __restrict__ Wv, const float* __restrict__ Wi,
    const float* __restrict__ Wf, const float* __restrict__ Wo,
    const float* __restrict__ b_i, const float* __restrict__ b_f,
    const float* __restrict__ b_o,
    float* __restrict__ oq, float* __restrict__ ok, float* __restrict__ ov,
    float* __restrict__ oi, float* __restrict__ of, float* __restrict__ oo)
{
    const int lane   = threadIdx.x & 31;
    const int waveId = threadIdx.x >> 5;
    const int tile   = blockIdx.x * 4 + waveId;     // 0 .. 24575

    const int matIdx = tile / (512 * 8);            // 0..5 : q,k,v,i,f,o
    const int rem    = tile % (512 * 8);
    const int m0     = (rem >> 3) << 4;
    const int n0     = (rem & 7) << 4;

    const float* W;
    switch (matIdx) {
        case 0: W = Wq; break;
        case 1: W = Wk; break;
        case 2: W = Wv; break;
        case 3: W = Wi; break;
        case 4: W = Wf; break;
        default: W = Wo; break;
    }

    const int lm = lane & 15;
    const int lk = (lane >> 4) << 1;

    const float* aptr = x + (m0 + lm) * HH + lk;
    const float* bptr = W + (n0 + lm) * HH + lk;

    v8f acc = {};
    #pragma unroll
    for (int kb = 0; kb < HH; kb += 4) {
        v2f a = *(const v2f*)(aptr + kb);
        v2f b = *(const v2f*)(bptr + kb);
        acc = __builtin_amdgcn_wmma_f32_16x16x4_f32(false, a, false, b,
                                                    (short)0, acc, false, false);
    }

    const int col = n0 + lm;
    float bias = 0.0f;
    if (matIdx == 3) bias = b_i[col];
    else if (matIdx == 4) bias = b_f[col];
    else if (matIdx == 5) bias = b_o[col];

    float* outp;
    switch (matIdx) {
        case 0: outp = oq; break;
        case 1: outp = ok; break;
        case 2: outp = ov; break;
        case 3: outp = oi; break;
        case 4: outp = of; break;
        default: outp = oo; break;
    }

    const int rowBase = m0 + ((lane >> 4) << 3);
    #pragma unroll
    for (int v = 0; v < 8; ++v) {
        float val = acc[v];
        if (matIdx <= 1)       val *= SCALE;
        else if (matIdx == 3)  val = expf(val + bias);
        else if (matIdx >= 4)  val = 1.0f / (1.0f + expf(-(val + bias)));
        outp[(rowBase + v) * HH + col] = val;
    }
}

// ---------------------------------------------------------------------------
// Kernel 2: chunkwise mLSTM recurrence.  One WG per batch, 128 thr (4 waves).
// ---------------------------------------------------------------------------
#define CSTR 132                         // C / Q / K / P / M row stride
#define TSTR 20                          // Kt / Wc row stride
#define C_OFF   0
#define Q_OFF   (C_OFF + 128 * CSTR)     // Q[s][c]  16x132
#define K_OFF   (Q_OFF + 16 * CSTR)      // K[s][c]  16x132
#define KT_OFF  (K_OFF + 16 * CSTR)      // Kt[c][s] 128x20
#define WC_OFF  (KT_OFF + 128 * TSTR)    // Wc[r][u] 128x20
#define P_OFF   (WC_OFF + 128 * TSTR)    // P^T[s][r] 16x132
#define M_OFF   (P_OFF + 16 * CSTR)      // n*q products [s][r] 16x132
#define SP_OFF  (M_OFF + 16 * CSTR)      // S partials [4][16][17]
#define S_OFF   (SP_OFF + 4 * 16 * 17)   // S [u][s] 16x17
#define G15_OFF (S_OFF + 16 * 17)        // G15[r] 128
#define NPS_OFF (G15_OFF + 128)          // nq partials [16][8]
#define NQ_OFF  (NPS_OFF + 128)          // nq[s] 16
#define SMEM_FLOATS (NQ_OFF + 16)

__global__ __launch_bounds__(128, 1) void rec_kernel(
    const float* __restrict__ qb, const float* __restrict__ kb,
    const float* __restrict__ vb, const float* __restrict__ ib,
    const float* __restrict__ fb, const float* __restrict__ ob,
    float* __restrict__ hb)
{
    extern __shared__ float sm[];
    const int b    = blockIdx.x;
    const int r    = threadIdx.x;        // row / thread 0..127
    const int lane = r & 31;
    const int wave = r >> 5;
    const int lm   = lane & 15;          // frag row/col index
    const int lk   = (lane >> 4) << 1;   // frag K sub-offset (0 or 2)

    // zero C state
    #pragma unroll 4
    for (int j = 0; j < CSTR; ++j) sm[C_OFF + r * CSTR + j] = 0.0f;

    float nprev = 0.0f;
    const long bbase = (long)b * TT * HH + r;

    // -------- software pipeline: prefetch chunk 0 gates into VGPRs --------
    float pk[LCH], pq[LCH], pi[LCH], pf[LCH], pv[LCH], po[LCH];
    #pragma unroll
    for (int s = 0; s < LCH; ++s) {
        const long idx = bbase + (long)s * HH;
        pk[s] = kb[idx]; pq[s] = qb[idx]; pi[s] = ib[idx];
        pf[s] = fb[idx]; pv[s] = vb[idx]; po[s] = ob[idx];
    }

    for (int ch = 0; ch < NCH; ++ch) {
        const long gbase = bbase + (long)(ch * LCH) * HH;

        // ---------------- phase 0: consume prefetched gates ----------------
        float G[LCH], Wg[LCH], ovv[LCH];
        {
            float g = 1.0f;
            #pragma unroll
            for (int s = 0; s < LCH; ++s) { g *= pf[s]; G[s] = g; ovv[s] = po[s]; }
            #pragma unroll
            for (int s = 0; s < LCH; ++s) Wg[s] = pi[s] * pv[s] / G[s];   // i*v/G_u
            const float g15 = G[LCH - 1];
            // n recurrence + n*q products
            float np = nprev;
            #pragma unroll
            for (int s = 0; s < LCH; ++s) {
                np = fmaf(pf[s], np, pi[s] * pk[s]);
                sm[M_OFF + s * CSTR + r] = np * pq[s];
            }
            nprev = np;
            // stage chunk matrices
            #pragma unroll
            for (int s = 0; s < LCH; ++s) {
                sm[Q_OFF + s * CSTR + r]  = pq[s];
                sm[K_OFF + s * CSTR + r]  = pk[s];
                sm[KT_OFF + r * TSTR + s] = pk[s];
                sm[WC_OFF + r * TSTR + s] = g15 * Wg[s];
            }
            sm[G15_OFF + r] = g15;
        }
        // -------- issue next chunk's gate loads (hidden behind phases 1-4) --
        if (ch + 1 < NCH) {
            const long nbase = bbase + (long)((ch + 1) * LCH) * HH;
            #pragma unroll
            for (int s = 0; s < LCH; ++s) {
                const long idx = nbase + (long)s * HH;
                pk[s] = kb[idx]; pq[s] = qb[idx]; pi[s] = ib[idx];
                pf[s] = fb[idx]; pv[s] = vb[idx]; po[s] = ob[idx];
            }
        }
        __syncthreads();   // A

        // ---------------- phase 1: nq part-1  +  P = C0 @ Q^T --------------
        {   // nq partial: thread j sums 16 of row s = j>>3, segment j&7
            const int s = r >> 3, seg = r & 7;
            float ps = 0.0f;
            #pragma unroll
            for (int e = 0; e < 16; ++e) ps += sm[M_OFF + s * CSTR + seg * 16 + e];
            sm[NPS_OFF + s * 8 + seg] = ps;
        }
        {   // P^T[s][r]: A = Q (16x128), B = C0^T tiles; 2 r-tiles per wave
            v2f av[32];
            #pragma unroll
            for (int kk = 0; kk < 32; ++kk)
                av[kk] = *(const v2f*)&sm[Q_OFF + lm * CSTR + kk * 4 + lk];
            #pragma unroll
            for (int tile = 0; tile < 2; ++tile) {
                const int n0p = (2 * wave + tile) << 4;
                v8f acc = {};
                #pragma unroll
                for (int kk = 0; kk < 32; ++kk) {
                    v2f bbf = *(const v2f*)&sm[C_OFF + (n0p + lm) * CSTR + kk * 4 + lk];
                    acc = __builtin_amdgcn_wmma_f32_16x16x4_f32(false, av[kk], false, bbf,
                                                                (short)0, acc, false, false);
                }
                #pragma unroll
                for (int v = 0; v < 8; ++v)
                    sm[P_OFF + (v + 8 * (lane >> 4)) * CSTR + n0p + lm] = acc[v];
            }
        }
        __syncthreads();   // B

        // ---------------- phase 2: nq final, S partials, C update ----------
        if (r < 16) {
            float t = 0.0f;
            #pragma unroll
            for (int e = 0; e < 8; ++e) t += sm[NPS_OFF + r * 8 + e];
            sm[NQ_OFF + r] = t;
        }
        {   // S partial over this wave's K-quarter: A = K rows, B = Q rows
            const int kbase = wave * 32;
            v8f sacc = {};
            #pragma unroll
            for (int kk = 0; kk < 8; ++kk) {
                const int c0 = kbase + kk * 4 + lk;
                v2f a  = *(const v2f*)&sm[K_OFF + lm * CSTR + c0];
                v2f bq = *(const v2f*)&sm[Q_OFF + lm * CSTR + c0];
                sacc = __builtin_amdgcn_wmma_f32_16x16x4_f32(false, a, false, bq,
                                                             (short)0, sacc, false, false);
            }
            #pragma unroll
            for (int v = 0; v < 8; ++v)
                sm[SP_OFF + wave * 272 + (v + 8 * (lane >> 4)) * 17 + lm] = sacc[v];
        }
        {   // C' = diag(G15) C0 + Wc @ K : wave handles row-tiles 2w, 2w+1
            const int mt0 = wave * 2;
            v2f afr[2][4];
            #pragma unroll
            for (int mi = 0; mi < 2; ++mi)
                #pragma unroll
                for (int kk = 0; kk < 4; ++kk)
                    afr[mi][kk] = *(const v2f*)&sm[WC_OFF + ((mt0 + mi) * 16 + lm) * TSTR + kk * 4 + lk];
            #pragma unroll
            for (int nt = 0; nt < 8; ++nt) {
                v2f bfr[4];
                #pragma unroll
                for (int kk = 0; kk < 4; ++kk)
                    bfr[kk] = *(const v2f*)&sm[KT_OFF + (nt * 16 + lm) * TSTR + kk * 4 + lk];
                #pragma unroll
                for (int mi = 0; mi < 2; ++mi) {
                    v8f dacc = {};
                    #pragma unroll
                    for (int kk = 0; kk < 4; ++kk)
                        dacc = __builtin_amdgcn_wmma_f32_16x16x4_f32(false, afr[mi][kk], false, bfr[kk],
                                                                     (short)0, dacc, false, false);
                    const int m0r = (mt0 + mi) * 16;
                    #pragma unroll
                    for (int v = 0; v < 8; ++v) {
                        const int row = m0r + v + 8 * (lane >> 4);
                        const int ci  = C_OFF + row * CSTR + nt * 16 + lm;
                        const float g = sm[G15_OFF + row];
                        sm[ci] = fmaf(g, sm[ci], dacc[v]);
                    }
                }
            }
        }
        __syncthreads();   // C

        // ---------------- phase 3: finalize S ------------------------------
        #pragma unroll
        for (int p = 0; p < 2; ++p) {
            const int idx = r * 2 + p;
            const int u = idx >> 4, sc = idx & 15;
            float ssum = sm[SP_OFF + 0 * 272 + u * 17 + sc]
                       + sm[SP_OFF + 1 * 272 + u * 17 + sc]
                       + sm[SP_OFF + 2 * 272 + u * 17 + sc]
                       + sm[SP_OFF + 3 * 272 + u * 17 + sc];
            sm[S_OFF + u * 17 + sc] = ssum;
        }
        __syncthreads();   // D

        // ---------------- phase 4: combine + output h ----------------------
        #pragma unroll
        for (int s = 0; s < LCH; ++s) {
            float acc = sm[P_OFF + s * CSTR + r];
            #pragma unroll
            for (int u = 0; u < LCH; ++u)
                if (u <= s) acc = fmaf(Wg[u], sm[S_OFF + u * 17 + s], acc);
            const float ret = G[s] * acc;
            const float nrm = fmaxf(fabsf(sm[NQ_OFF + s]), 1.0f);
            hb[gbase + (long)s * HH] = ovv[s] * ret / nrm;
        }
        __syncthreads();   // protects LDS re-staging of next chunk
    }
}

// ---------------------------------------------------------------------------
// Kernel 3: residual + LayerNorm.
// ---------------------------------------------------------------------------
__global__ __launch_bounds__(256) void ln_kernel(
    const float* __restrict__ hb, const float* __restrict__ x,
    const float* __restrict__ lnw, const float* __restrict__ lnb,
    float* __restrict__ out)
{
    const int lane = threadIdx.x & 31;
    const int wave = threadIdx.x >> 5;
    const long row = (long)blockIdx.x * 8 + wave;
    const long rb  = row * HH;

    float y[4];
    float s = 0.0f, s2 = 0.0f;
    #pragma unroll
    for (int j = 0; j < 4; ++j) {
        const int c = lane + 32 * j;
        y[j] = hb[rb + c] + x[rb + c];
        s  += y[j];
        s2 += y[j] * y[j];
    }
    #pragma unroll
    for (int off = 16; off >= 1; off >>= 1) {
        s  += __shfl_xor(s,  off, 32);
        s2 += __shfl_xor(s2, off, 32);
    }
    const float mu  = s * (1.0f / HH);
    const float var = s2 * (1.0f / HH) - mu * mu;
    const float rs  = rsqrtf(var + LN_EPS);

    #pragma unroll
    for (int j = 0; j < 4; ++j) {
        const int c = lane + 32 * j;
        out[rb + c] = (y[j] - mu) * rs * lnw[c] + lnb[c];
    }
}

// ---------------------------------------------------------------------------
extern "C" void kernel_launch(void* const* d_in, const int* in_sizes, int n_in,
                              void* d_out, int out_size, void* d_ws, size_t ws_size,
                              hipStream_t stream)
{
    const float* x   = (const float*)d_in[0];
    const float* Wq  = (const float*)d_in[1];
    const float* Wk  = (const float*)d_in[2];
    const float* Wv  = (const float*)d_in[3];
    const float* Wi  = (const float*)d_in[4];
    const float* Wf  = (const float*)d_in[5];
    const float* Wo  = (const float*)d_in[6];
    const float* b_i = (const float*)d_in[7];
    const float* b_f = (const float*)d_in[8];
    const float* b_o = (const float*)d_in[9];
    const float* lnw = (const float*)d_in[10];
    const float* lnb = (const float*)d_in[11];
    float* out = (float*)d_out;

    float* ws = (float*)d_ws;
    float* q  = ws + 0L * NEL;
    float* k  = ws + 1L * NEL;
    float* v  = ws + 2L * NEL;
    float* ig = ws + 3L * NEL;
    float* fg = ws + 4L * NEL;
    float* og = ws + 5L * NEL;
    float* h  = ws + 6L * NEL;

    proj_kernel<<<6144, 128, 0, stream>>>(x, Wq, Wk, Wv, Wi, Wf, Wo,
                                          b_i, b_f, b_o,
                                          q, k, v, ig, fg, og);
    rec_kernel<<<BB, 128, SMEM_FLOATS * sizeof(float), stream>>>(q, k, v, ig, fg, og, h);
    ln_kernel<<<MM / 8, 256, 0, stream>>>(h, x, lnw, lnb, out);
}